// TimeSeriesModel_13271448944756
// MI455X (gfx1250) — compile-verified
//
#include <hip/hip_runtime.h>

typedef __attribute__((ext_vector_type(2))) float v2f;
typedef __attribute__((ext_vector_type(8))) float v8f;

#define BATCH 256
#define NPIPE 512
#define TLEN  250
#define OVL   25

// ---------------------------------------------------------------------------
// Kernel A: xm[b,t] = (1/512) * sum_p x[b,p,t]
// One wave (32 lanes) per (batch, 16-wide time tile).
// A = ones(16x4) f32, B = 4x16 tile of pipe rows -> C row M=0 accumulates the
// pipe sum across 128 chained V_WMMA_F32_16X16X4_F32 issues.
// B layout (32-bit, 4x16): VGPR0 = {K=0 lanes 0-15, K=2 lanes 16-31},
//                          VGPR1 = {K=1 lanes 0-15, K=3 lanes 16-31}.
// C layout: VGPR0 lanes 0-15 = row M=0, N=lane.
// ---------------------------------------------------------------------------
__global__ __launch_bounds__(32) void mean_wmma_kernel(const float* __restrict__ x,
                                                       float* __restrict__ xm) {
  const int b    = blockIdx.y;
  const int t0   = blockIdx.x * 16;
  const int lane = threadIdx.x;      // 0..31, full wave, EXEC all ones
  const int half = lane >> 4;        // selects K=0/1 vs K=2/3 rows
  const int tl   = lane & 15;
  const int t    = t0 + tl;
  const int tc   = (t < TLEN) ? t : (TLEN - 1);  // clamp tail; lanes discarded at store

  const float* __restrict__ xb = x + (size_t)b * NPIPE * TLEN;

  v2f a; a.x = 1.0f; a.y = 1.0f;     // A = ones(16x4): every row of C = pipe sum
  v8f c = {};

  for (int p = 0; p < NPIPE; p += 4) {
    const float* r = xb + (size_t)(p + 2 * half) * TLEN + tc;
    v2f bv;
    bv.x = r[0];        // pipe row p + 2*half
    bv.y = r[TLEN];     // pipe row p + 2*half + 1
    c = __builtin_amdgcn_wmma_f32_16x16x4_f32(false, a, false, bv,
                                              (short)0, c, false, false);
  }

  if (lane < 16 && t < TLEN)
    xm[(size_t)b * TLEN + t] = c[0] * (1.0f / (float)NPIPE);
}

// ---------------------------------------------------------------------------
// Kernel B: squeeze-excite weight vector per batch (tiny MLPs, VALU).
// weights[b, 0:250] built from 9 overlapping 50-windows + one 25-window.
// ---------------------------------------------------------------------------
__device__ __forceinline__ float sigmoidf_(float z) {
  return 1.0f / (1.0f + __expf(-z));
}

__global__ __launch_bounds__(256) void se_weights_kernel(
    const float* __restrict__ xm,
    const float* __restrict__ w1_50, const float* __restrict__ b1_50,
    const float* __restrict__ w2_50, const float* __restrict__ b2_50,
    const float* __restrict__ w1_25, const float* __restrict__ b1_25,
    const float* __restrict__ w2_25, const float* __restrict__ b2_25,
    float* __restrict__ weights) {
  __shared__ float xs[TLEN];
  __shared__ float hs[19];  // hs[i*2+r] = relu hidden (window i, unit r); hs[18] = h25

  const int b   = blockIdx.x;
  const int tid = threadIdx.x;

  if (tid < TLEN) xs[tid] = xm[(size_t)b * TLEN + tid];
  __syncthreads();

  if (tid < 18) {                 // 9 windows x 2 hidden units, 50-long dots
    const int i = tid >> 1, r = tid & 1;
    const float* w   = w1_50 + r * 50;
    const float* win = xs + i * OVL;
    float acc = b1_50[r];
    #pragma unroll
    for (int k = 0; k < 50; ++k) acc = fmaf(win[k], w[k], acc);
    hs[tid] = fmaxf(acc, 0.0f);
  } else if (tid == 18) {         // 25-window hidden unit
    float acc = b1_25[0];
    #pragma unroll
    for (int k = 0; k < OVL; ++k) acc = fmaf(xs[TLEN - OVL + k], w1_25[k], acc);
    hs[18] = fmaxf(acc, 0.0f);
  }
  __syncthreads();

  if (tid < TLEN) {
    const int seg = tid / OVL;    // 0..9
    const int pos = tid % OVL;

    // w50[i][j] = sigmoid(h[i,0]*w2_50[j,0] + h[i,1]*w2_50[j,1] + b2_50[j])
    auto sig50 = [&](int i, int j) -> float {
      return sigmoidf_(fmaf(hs[i * 2], w2_50[j * 2],
                       fmaf(hs[i * 2 + 1], w2_50[j * 2 + 1], b2_50[j])));
    };

    float wv;
    if (seg == 0) {
      wv = sig50(0, pos);                                   // L[0]
    } else if (seg < 9) {
      wv = 0.5f * (sig50(seg - 1, OVL + pos) + sig50(seg, pos));  // (R[k-1]+L[k])/2
    } else {
      float w25v = sigmoidf_(fmaf(hs[18], w2_25[pos], b2_25[pos]));
      wv = 0.5f * (sig50(8, OVL + pos) + w25v);             // (R[8]+w25)/2
    }
    weights[(size_t)b * TLEN + tid] = wv;
  }
}

// ---------------------------------------------------------------------------
// Kernel C: out[b,p,t] = x[b,p,t] * weights[b,t]
// One 128-thread block per (pipe, batch) row; float2 vectorized (rows are
// 1000 B -> 8-byte aligned). Weight rows are reused by 512 blocks -> L2-hot.
// ---------------------------------------------------------------------------
__global__ __launch_bounds__(128) void scale_kernel(const float* __restrict__ x,
                                                    const float* __restrict__ weights,
                                                    float* __restrict__ out) {
  const int p = blockIdx.x, b = blockIdx.y, tid = threadIdx.x;
  const size_t row = ((size_t)b * NPIPE + p) * TLEN;

  const float2* __restrict__ xr = (const float2*)(x + row);
  const float2* __restrict__ wr = (const float2*)(weights + (size_t)b * TLEN);
  float2* __restrict__ outr     = (float2*)(out + row);

  if (tid < TLEN / 2) {
    float2 xv = xr[tid];
    float2 wv = wr[tid];
    outr[tid] = make_float2(xv.x * wv.x, xv.y * wv.y);
  }
}

// ---------------------------------------------------------------------------
extern "C" void kernel_launch(void* const* d_in, const int* in_sizes, int n_in,
                              void* d_out, int out_size, void* d_ws, size_t ws_size,
                              hipStream_t stream) {
  const float* x     = (const float*)d_in[0];
  const float* w1_50 = (const float*)d_in[1];
  const float* b1_50 = (const float*)d_in[2];
  const float* w2_50 = (const float*)d_in[3];
  const float* b2_50 = (const float*)d_in[4];
  const float* w1_25 = (const float*)d_in[5];
  const float* b1_25 = (const float*)d_in[6];
  const float* w2_25 = (const float*)d_in[7];
  const float* b2_25 = (const float*)d_in[8];

  float* out     = (float*)d_out;
  float* xm      = (float*)d_ws;                 // 256*250 f32
  float* weights = xm + (size_t)BATCH * TLEN;    // 256*250 f32

  hipLaunchKernelGGL(mean_wmma_kernel, dim3(16, BATCH), dim3(32), 0, stream, x, xm);
  hipLaunchKernelGGL(se_weights_kernel, dim3(BATCH), dim3(256), 0, stream,
                     xm, w1_50, b1_50, w2_50, b2_50, w1_25, b1_25, w2_25, b2_25,
                     weights);
  hipLaunchKernelGGL(scale_kernel, dim3(NPIPE, BATCH), dim3(128), 0, stream,
                     x, weights, out);
}